// MILNCE_44006234915388
// MI455X (gfx1250) — compile-verified
//
#include <hip/hip_runtime.h>

// MIL-NCE / supervised-contrastive loss, fused on gfx1250.
// z = (fhat @ fhat^T)/tau via V_WMMA_F32_16X16X4_F32, logsumexp fused with a
// fixed shift (|z| <= 1/TAU = 10), per-row lse difference, deterministic sum.

typedef __attribute__((ext_vector_type(2))) float v2f;
typedef __attribute__((ext_vector_type(4))) float v4f;
typedef __attribute__((ext_vector_type(8))) float v8f;

constexpr int   D        = 256;
constexpr int   B_ROWS   = 4096;
constexpr int   N        = 8192;          // 2*B_ROWS
constexpr int   TILE     = 16;
constexpr int   NT       = N / TILE;      // 512 column tiles
constexpr int   LDS_STR  = 260;           // 260 % 64 == 4 -> conflict-free b64 reads
constexpr float INV_SQRT_TAU = 3.16227766016838f;  // 1/sqrt(0.1)
constexpr float ZMAX     = 10.0f;         // 1/TAU, fixed logsumexp shift

// ---------------------------------------------------------------------------
// Kernel 1: fhat[row] = f[row] / max(||f[row]||, eps) * (1/sqrt(tau))
// One block (256 threads) per row, D == 256.
// ---------------------------------------------------------------------------
__global__ __launch_bounds__(256)
void milnce_normalize(const float* __restrict__ feat,
                      const float* __restrict__ pos,
                      float* __restrict__ fhat) {
    const int row = blockIdx.x;           // 0..N-1
    const int t   = threadIdx.x;          // 0..255
    const float* src = (row < B_ROWS) ? (feat + (size_t)row * D)
                                      : (pos + (size_t)(row - B_ROWS) * D);
    const float x = src[t];

    // sum of squares: wave32 shfl reduce, then 8 partials via LDS
    float ss = x * x;
    #pragma unroll
    for (int off = 16; off > 0; off >>= 1) ss += __shfl_xor(ss, off, 32);

    __shared__ float red[8];
    if ((t & 31) == 0) red[t >> 5] = ss;
    __syncthreads();

    float tot = 0.0f;
    #pragma unroll
    for (int i = 0; i < 8; ++i) tot += red[i];

    const float scale = rsqrtf(fmaxf(tot, 1e-16f)) * INV_SQRT_TAU;
    fhat[(size_t)row * D + t] = x * scale;
}

// ---------------------------------------------------------------------------
// Kernel 2: fused  z-tile GEMM (WMMA f32 16x16x4) + masked exp accumulation.
// Block = 256 thr = 8 waves; wave w owns rows [blk*128 + w*16, +16).
// Iterates all 512 column tiles; B tiles double-buffered through LDS.
// ---------------------------------------------------------------------------
__global__ __launch_bounds__(256)
void milnce_main(const float* __restrict__ fhat,
                 const int*   __restrict__ labels,   // B_ROWS entries
                 float*       __restrict__ row_out)  // N entries: lse_all - lse_pos
{
    __shared__ __align__(16) float lds[2][TILE][LDS_STR];

    const int t    = threadIdx.x;
    const int wave = t >> 5;
    const int lane = t & 31;
    const int lh   = lane & 15;     // lane within 16-lane half
    const int half = lane >> 4;     // 0 or 1
    const int r0   = blockIdx.x * 128 + wave * 16;

    // ---- A fragments for the whole K=256 strip, kept in registers --------
    // Lane l<16 holds A[M=l][K=4k .. 4k+1]; lane l>=16 holds A[M=l-16][K=4k+2 .. 4k+3]
    const float* aptr = fhat + (size_t)(r0 + lh) * D + 2 * half;
    v2f areg[64];
    #pragma unroll
    for (int kk = 0; kk < 64; ++kk)
        areg[kk] = *(const v2f*)(aptr + kk * 4);

    // ---- row labels for this lane's 8 accumulator rows -------------------
    int labR[8];
    #pragma unroll
    for (int v = 0; v < 8; ++v)
        labR[v] = labels[(r0 + half * 8 + v) & (B_ROWS - 1)];

    float sAll[8], sPos[8];
    #pragma unroll
    for (int v = 0; v < 8; ++v) { sAll[v] = 0.0f; sPos[v] = 0.0f; }

    // ---- cooperative B-tile staging: thread t loads row t/16, 16 floats --
    const int ldRow = t >> 4;             // 0..15
    const int ldCol = (t & 15) * 16;      // 0,16,..,240
    v4f pf[4];
    {
        const float* src = fhat + (size_t)ldRow * D + ldCol;
        #pragma unroll
        for (int j = 0; j < 4; ++j) pf[j] = *(const v4f*)(src + j * 4);
        #pragma unroll
        for (int j = 0; j < 4; ++j)
            *(v4f*)&lds[0][ldRow][ldCol + j * 4] = pf[j];
    }
    __syncthreads();

    // per-lane LDS base for B fragments (same layout as A, rows = tile cols)
    for (int tile = 0; tile < NT; ++tile) {
        const int cur = tile & 1;
        const int c0  = tile * TILE;

        // prefetch next tile into registers (hides latency behind WMMAs)
        if (tile + 1 < NT) {
            const float* src = fhat + (size_t)(c0 + TILE + ldRow) * D + ldCol;
            #pragma unroll
            for (int j = 0; j < 4; ++j) pf[j] = *(const v4f*)(src + j * 4);
        }

        // ---- 16x16 z-tile: 64 chained f32 WMMAs over K=256 ---------------
        const float* bbase = &lds[cur][lh][2 * half];
        v8f c = {0.f, 0.f, 0.f, 0.f, 0.f, 0.f, 0.f, 0.f};
        #pragma unroll
        for (int kk = 0; kk < 64; ++kk) {
            v2f b = *(const v2f*)(bbase + kk * 4);
            c = __builtin_amdgcn_wmma_f32_16x16x4_f32(
                    /*neg_a=*/false, areg[kk],
                    /*neg_b=*/false, b,
                    /*c_mod=*/(short)0, c,
                    /*reuse_a=*/false, /*reuse_b=*/false);
        }

        // ---- epilogue: masked exp(z - 10) accumulation --------------------
        // C layout: vgpr v, lanes 0-15 -> M=v, N=lane; lanes 16-31 -> M=8+v.
        const int gc   = c0 + lh;
        const int labC = labels[gc & (B_ROWS - 1)];
        #pragma unroll
        for (int v = 0; v < 8; ++v) {
            const int gr = r0 + half * 8 + v;
            float e = __expf(c[v] - ZMAX);
            e = (gc == gr) ? 0.0f : e;              // drop diagonal
            sAll[v] += e;
            sPos[v] += (labC == labR[v]) ? e : 0.0f;
        }

        __syncthreads();
        if (tile + 1 < NT) {
            #pragma unroll
            for (int j = 0; j < 4; ++j)
                *(v4f*)&lds[cur ^ 1][ldRow][ldCol + j * 4] = pf[j];
        }
        __syncthreads();
    }

    // ---- reduce partial sums across the 16 lanes of each half ------------
    #pragma unroll
    for (int v = 0; v < 8; ++v) {
        #pragma unroll
        for (int off = 1; off < 16; off <<= 1) {
            sAll[v] += __shfl_xor(sAll[v], off, 32);
            sPos[v] += __shfl_xor(sPos[v], off, 32);
        }
    }
    if (lh == 0) {
        #pragma unroll
        for (int v = 0; v < 8; ++v)
            row_out[r0 + half * 8 + v] = __logf(sAll[v]) - __logf(sPos[v]);
        // (fixed shift ZMAX cancels in lse_all - lse_pos)
    }
}

// ---------------------------------------------------------------------------
// Kernel 3: deterministic scalar reduction of the N per-row differences.
// ---------------------------------------------------------------------------
__global__ __launch_bounds__(256)
void milnce_reduce(const float* __restrict__ row_out, float* __restrict__ out) {
    __shared__ float red[256];
    const int t = threadIdx.x;
    float s = 0.0f;
    for (int i = t; i < N; i += 256) s += row_out[i];  // fixed order
    red[t] = s;
    __syncthreads();
    for (int off = 128; off > 0; off >>= 1) {
        if (t < off) red[t] += red[t + off];
        __syncthreads();
    }
    if (t == 0) out[0] = red[0];
}

// ---------------------------------------------------------------------------
extern "C" void kernel_launch(void* const* d_in, const int* in_sizes, int n_in,
                              void* d_out, int out_size, void* d_ws, size_t ws_size,
                              hipStream_t stream) {
    const float* feat   = (const float*)d_in[0];   // [4096, 256] f32
    const float* pos    = (const float*)d_in[1];   // [4096, 256] f32
    const int*   labels = (const int*)d_in[2];     // [4096] int
    float* out = (float*)d_out;                    // scalar f32

    float* ws      = (float*)d_ws;
    float* row_out = ws;                           // N floats
    float* fhat    = ws + N;                       // N*D floats (8 MB)

    milnce_normalize<<<N, 256, 0, stream>>>(feat, pos, fhat);
    milnce_main<<<N / 128, 256, 0, stream>>>(fhat, labels, row_out);
    milnce_reduce<<<1, 256, 0, stream>>>(row_out, out);
}